// ARMAConv_80161269612936
// MI455X (gfx1250) — compile-verified
//
#include <hip/hip_runtime.h>

typedef __attribute__((ext_vector_type(2))) float v2f;
typedef __attribute__((ext_vector_type(8))) float v8f;

#define N_NODES 50000
#define N_EDGES 400000
#define IC 128
#define OC 128
#define FDIM 256                       // OC*WIDTH
#define N_TILES (FDIM / 16)            // 16 column tiles
#define MB 5                           // row tiles per wave (3125 = 5*625)
#define MACRO_M (MB * 16)              // 80 rows per wave
#define MACRO_TILES (N_NODES / MACRO_M)       // 625
#define WAVES_PER_BLOCK 8
#define GEMM_BLOCKS ((MACRO_TILES * N_TILES) / WAVES_PER_BLOCK)  // 1250

// ---------------- utility kernels ----------------

__global__ void zero_i32(int* __restrict__ p, int n) {
    int i = blockIdx.x * blockDim.x + threadIdx.x;
    if (i < n) p[i] = 0;
}

__global__ void zero_f32x4(float4* __restrict__ p, int n4) {
    int i = blockIdx.x * blockDim.x + threadIdx.x;
    if (i < n4) p[i] = make_float4(0.f, 0.f, 0.f, 0.f);
}

__global__ void degree_kernel(const int* __restrict__ dst, int* __restrict__ deg) {
    int i = blockIdx.x * blockDim.x + threadIdx.x;
    if (i < N_EDGES) atomicAdd(&deg[dst[i]], 1);
}

__global__ void norm_kernel(const int* __restrict__ deg, float* __restrict__ norm) {
    int i = blockIdx.x * blockDim.x + threadIdx.x;
    if (i < N_NODES) {
        int d = deg[i];
        norm[i] = (d > 0) ? rsqrtf((float)d) : 0.0f;
    }
}

__global__ void combine_weights(const float* __restrict__ Wp, const float* __restrict__ Wm,
                                float* __restrict__ Wc,
                                const float* __restrict__ bp, const float* __restrict__ bm,
                                float* __restrict__ bc) {
    int i = blockIdx.x * blockDim.x + threadIdx.x;
    if (i < IC * FDIM) Wc[i] = Wp[i] + Wm[i];
    if (i < FDIM)      bc[i] = bp[i] + bm[i];
}

// ---------------- GEMM 1: merge = X@Wm + bm ; out = relu(X@Wcomb + bcomb) ----------------
// One wave computes MB=5 row-tiles x 1 col-tile (80x16) so each B fetch feeds 5 WMMAs
// per weight matrix. K=128 via fp32 WMMA 16x16x4. EXEC stays all-ones (no masking).

__global__ void __launch_bounds__(256)
gemm_in(const float* __restrict__ A, const float* __restrict__ Wmerge,
        const float* __restrict__ Wcomb, const float* __restrict__ bmerge,
        const float* __restrict__ bcomb, float* __restrict__ merge,
        float* __restrict__ out) {
    const int wave = blockIdx.x * WAVES_PER_BLOCK + (threadIdx.x >> 5);
    const int mt = wave >> 4;               // macro row tile: 0..624
    const int tn = wave & (N_TILES - 1);    // col tile: 0..15
    const int lane = threadIdx.x & 31;
    const int half = lane >> 4;             // 0: K pair {k,k+1}, 1: {k+2,k+3}
    const int l = lane & 15;
    const int col = tn * 16 + l;            // B/C column for this lane
    const int row0 = mt * MACRO_M + l;      // first A row for this lane

    const float* a0 = A + (size_t)row0 * IC;

    v8f accM[MB];
    v8f accC[MB];
    const v8f vzero = {};
#pragma unroll
    for (int r = 0; r < MB; ++r) { accM[r] = vzero; accC[r] = vzero; }

#pragma unroll
    for (int k = 0; k < IC; k += 4) {
        const int kb = k + 2 * half;
        v2f bM, bC;
        bM.x = Wmerge[(size_t)kb * FDIM + col];
        bM.y = Wmerge[(size_t)(kb + 1) * FDIM + col];
        bC.x = Wcomb[(size_t)kb * FDIM + col];
        bC.y = Wcomb[(size_t)(kb + 1) * FDIM + col];
#pragma unroll
        for (int r = 0; r < MB; ++r) {
            v2f a = *(const v2f*)(a0 + (size_t)r * 16 * IC + kb);   // aligned float2
            accM[r] = __builtin_amdgcn_wmma_f32_16x16x4_f32(false, a, false, bM, (short)0, accM[r], false, false);
            accC[r] = __builtin_amdgcn_wmma_f32_16x16x4_f32(false, a, false, bC, (short)0, accC[r], false, false);
        }
    }
    const float bm = bmerge[col];
    const float bc = bcomb[col];
#pragma unroll
    for (int r = 0; r < MB; ++r) {
#pragma unroll
        for (int j = 0; j < 8; ++j) {
            const int rc = mt * MACRO_M + r * 16 + j + 8 * half;  // lanes 16-31 hold M=j+8
            const size_t idx = (size_t)rc * FDIM + col;
            merge[idx] = accM[r][j] + bm;
            const float o = accC[r][j] + bc;
            out[idx] = o > 0.f ? o : 0.f;
        }
    }
}

// ---------------- edge scatter: agg[dst] += norm[src] * x[src] ----------------
// 64 threads per edge, float4 gather + native fp32 atomics; the 51 MB agg buffer
// is L2-resident (192 MB L2) so atomics complete at L2 speed.

__global__ void __launch_bounds__(256)
scatter_kernel(const int* __restrict__ src, const int* __restrict__ dst,
               const float* __restrict__ norm, const float* __restrict__ x,
               float* __restrict__ agg) {
    const int t = blockIdx.x * blockDim.x + threadIdx.x;
    const int e = t >> 6;
    if (e >= N_EDGES) return;
    const int c = (t & 63) << 2;
    const int s = src[e];
    const int d = dst[e];
    const float ns = norm[s];
    const float4 v = *(const float4*)(x + (size_t)s * FDIM + c);
    float* p = agg + (size_t)d * FDIM + c;
    unsafeAtomicAdd(p + 0, v.x * ns);
    unsafeAtomicAdd(p + 1, v.y * ns);
    unsafeAtomicAdd(p + 2, v.z * ns);
    unsafeAtomicAdd(p + 3, v.w * ns);
}

// ---------------- GEMM 2 (per width w): out = relu((norm*agg_w) @ Wconv[w]^T + bconv + merge) ----------------
// B[k=i][n=o] = Wconv[w][o][i] is contiguous in k -> float2 B loads; norm scaling fused into A.

__global__ void __launch_bounds__(256)
gemm_conv(const float* __restrict__ agg, const float* __restrict__ norm,
          const float* __restrict__ Wconv, const float* __restrict__ bconv,
          const float* __restrict__ merge, float* __restrict__ out) {
    const int wave = blockIdx.x * WAVES_PER_BLOCK + (threadIdx.x >> 5);
    const int mt = wave >> 4;
    const int tn = wave & (N_TILES - 1);
    const int lane = threadIdx.x & 31;
    const int half = lane >> 4;
    const int l = lane & 15;
    const int col = tn * 16 + l;            // flat output column in [0,256)
    const int w = tn >> 3;                  // width block
    const int o = col & (OC - 1);
    const int row0 = mt * MACRO_M + l;

    const float* a0 = agg + (size_t)row0 * FDIM + w * OC;
    const float* brow = Wconv + (size_t)w * OC * OC + (size_t)o * OC;   // W_conv[w][o][*]

    float nr[MB];
#pragma unroll
    for (int r = 0; r < MB; ++r) nr[r] = norm[row0 + r * 16];

    v8f acc[MB];
    const v8f vzero = {};
#pragma unroll
    for (int r = 0; r < MB; ++r) acc[r] = vzero;

#pragma unroll
    for (int k = 0; k < OC; k += 4) {
        const int kb = k + 2 * half;
        v2f b = *(const v2f*)(brow + kb);
#pragma unroll
        for (int r = 0; r < MB; ++r) {
            v2f a = *(const v2f*)(a0 + (size_t)r * 16 * FDIM + kb);
            a.x *= nr[r];
            a.y *= nr[r];
            acc[r] = __builtin_amdgcn_wmma_f32_16x16x4_f32(false, a, false, b, (short)0, acc[r], false, false);
        }
    }
    const float bb = bconv[col];
#pragma unroll
    for (int r = 0; r < MB; ++r) {
#pragma unroll
        for (int j = 0; j < 8; ++j) {
            const int rc = mt * MACRO_M + r * 16 + j + 8 * half;
            const size_t idx = (size_t)rc * FDIM + col;
            const float v = acc[r][j] + bb + merge[idx];
            out[idx] = v > 0.f ? v : 0.f;
        }
    }
}

// ---------------- final pair-mean: y[n,c] = 0.5*(out[n,2c] + out[n,2c+1]) ----------------

__global__ void final_kernel(const float* __restrict__ out, float* __restrict__ y) {
    const int i = blockIdx.x * blockDim.x + threadIdx.x;
    if (i >= N_NODES * OC) return;
    const int n = i >> 7;
    const int c = i & (OC - 1);
    const float* p = out + (size_t)n * FDIM + 2 * c;
    y[i] = 0.5f * (p[0] + p[1]);
}

// ---------------- host ----------------

extern "C" void kernel_launch(void* const* d_in, const int* in_sizes, int n_in,
                              void* d_out, int out_size, void* d_ws, size_t ws_size,
                              hipStream_t stream) {
    (void)in_sizes; (void)n_in; (void)out_size; (void)ws_size;
    const float* node  = (const float*)d_in[0];
    const int*   ei    = (const int*)d_in[1];
    const int*   src   = ei;
    const int*   dstp  = ei + N_EDGES;
    const float* Wpre  = (const float*)d_in[2];
    const float* bpre  = (const float*)d_in[3];
    const float* Wmer  = (const float*)d_in[4];
    const float* bmer  = (const float*)d_in[5];
    const float* Wconv = (const float*)d_in[6];
    const float* bconv = (const float*)d_in[7];
    float* y = (float*)d_out;

    char* ws = (char*)d_ws;
    size_t off = 0;
    auto alloc = [&](size_t bytes) -> void* {
        void* p = ws + off;
        off += (bytes + 255) & ~(size_t)255;
        return p;
    };
    float* merge = (float*)alloc(sizeof(float) * (size_t)N_NODES * FDIM);
    float* outb  = (float*)alloc(sizeof(float) * (size_t)N_NODES * FDIM);
    float* agg   = (float*)alloc(sizeof(float) * (size_t)N_NODES * FDIM);
    float* norm  = (float*)alloc(sizeof(float) * N_NODES);
    int*   deg   = (int*)alloc(sizeof(int) * N_NODES);
    float* Wcomb = (float*)alloc(sizeof(float) * IC * FDIM);
    float* bcomb = (float*)alloc(sizeof(float) * FDIM);

    zero_i32<<<(N_NODES + 255) / 256, 256, 0, stream>>>(deg, N_NODES);
    degree_kernel<<<(N_EDGES + 255) / 256, 256, 0, stream>>>(dstp, deg);
    norm_kernel<<<(N_NODES + 255) / 256, 256, 0, stream>>>(deg, norm);
    combine_weights<<<(IC * FDIM + 255) / 256, 256, 0, stream>>>(Wpre, Wmer, Wcomb, bpre, bmer, bcomb);

    gemm_in<<<GEMM_BLOCKS, 256, 0, stream>>>(node, Wmer, Wcomb, bmer, bcomb, merge, outb);

    const int n4 = (N_NODES * FDIM) / 4;
    for (int it = 0; it < 2; ++it) {
        zero_f32x4<<<(n4 + 255) / 256, 256, 0, stream>>>((float4*)agg, n4);
        scatter_kernel<<<(N_EDGES * 64) / 256, 256, 0, stream>>>(src, dstp, norm, outb, agg);
        gemm_conv<<<GEMM_BLOCKS, 256, 0, stream>>>(agg, norm, Wconv, bconv, merge, outb);
    }

    final_kernel<<<(N_NODES * OC + 255) / 256, 256, 0, stream>>>(outb, y);
}